// RNN_13924283974382
// MI455X (gfx1250) — compile-verified
//
#include <hip/hip_runtime.h>
#include <hip/hip_bf16.h>
#include <stddef.h>
#include <stdint.h>

// ---------------------------------------------------------------------------
// Tree-RNN on MI455X (gfx1250), wave32, f32 WMMA 16x16x4.
//
// Roofline: dominated by one pass over leaf_seqs (128 MiB) -> ~5.8us @ 23.3TB/s.
// 6.7 GFLOP total => ~1.2 TFLOPS needed, far under the f32 WMMA ceiling, so we
// keep full f32 precision (matches the f32 reference) and optimize data motion.
// New this round: A-panel staging uses GLOBAL_LOAD_ASYNC_TO_LDS_B128
// (ASYNCcnt-tracked direct global->LDS DMA, no VGPR round trip) with explicit
// double buffering.
// ---------------------------------------------------------------------------

typedef __attribute__((ext_vector_type(2))) float v2f;
typedef __attribute__((ext_vector_type(8))) float v8f;

#define SEQ        4096
#define EMB        100
#define NLEAF      8192
#define NTILE      8            // 8 N-tiles of 16 (100 padded to 128; tile 7 zeros)
#define KLVL       200          // concat(left,right) width
#define KSTEPS_LVL (KLVL / 4)   // 50 WMMA k-steps per level row-block
#define LDSTR      68           // LDS row stride (floats): conflict-free frag reads

// Per-lane async copy: 16 bytes global -> LDS. VDST holds the LDS byte offset
// (low 32 bits of the generic pointer == LDS offset per the aperture mapping).
__device__ __forceinline__ void async_copy_b128(uint32_t lds_off, const float* gptr) {
    asm volatile("global_load_async_to_lds_b128 %0, %1, off"
                 :: "v"(lds_off), "v"(gptr) : "memory");
}

// ---------------------------------------------------------------------------
// Pre-swizzle a [nOut x kDim] row-major weight (used as B = W.T) into the
// WMMA 16x16x4 B-fragment layout: frag(kstep, tile) = 32 lanes x 2 floats.
//   lane l: c = (l<16)?0:2 ; n = tile*16 + (l&15)
//   frag[l][j] = W[n][kstep*4 + c + j]   (zero-padded for n >= nOut)
// ---------------------------------------------------------------------------
__global__ void swz_kernel(const float* __restrict__ Wsrc, float* __restrict__ dst,
                           int nOut, int kDim, int nFrags) {
    int t    = blockIdx.x * blockDim.x + threadIdx.x;
    int lane = t & 31;
    int frag = t >> 5;
    if (frag >= nFrags) return;
    int tile  = frag & (NTILE - 1);
    int kstep = frag >> 3;               // NTILE == 8
    int c = (lane < 16) ? 0 : 2;
    int n = tile * 16 + (lane & 15);
    int k = kstep * 4 + c;
    float x0 = 0.f, x1 = 0.f;
    if (n < nOut) {
        x0 = Wsrc[(size_t)n * kDim + k];
        x1 = Wsrc[(size_t)n * kDim + k + 1];
    }
    dst[(size_t)frag * 64 + lane * 2 + 0] = x0;
    dst[(size_t)frag * 64 + lane * 2 + 1] = x1;
}

// ---------------------------------------------------------------------------
// Leaf embedding: out[8192 x 100] = relu(A[8192 x 4096] @ We.T + be)
// One block = 16 output rows x 128 cols (8 waves, one 16x16 N-tile each).
// A double-buffered in LDS via async global->LDS b128 copies; B fragments
// streamed from the pre-swizzled L2-resident buffer (1 coalesced b64 / lane).
// ---------------------------------------------------------------------------
__global__ __launch_bounds__(256) void embed_gemm(const float* __restrict__ A,
                                                  const float* __restrict__ Bswz,
                                                  const float* __restrict__ bias,
                                                  float* __restrict__ out) {
    __shared__ float ldsA[2][16 * LDSTR];
    const int tid   = threadIdx.x;
    const int lane  = tid & 31;
    const int wave  = tid >> 5;          // == N-tile index (0..7)
    const int row16 = lane & 15;
    const int chalf = (lane < 16) ? 0 : 2;
    const int m0    = blockIdx.x * 16;

    const int ldr = tid >> 4;            // 0..15 : row staged by this thread
    const int ldk = (tid & 15) * 4;      // 0..60 : float4 column
    const float* gsrc = A + (size_t)m0 * SEQ + (size_t)ldr * SEQ + ldk;
    const uint32_t off0 = (uint32_t)(uintptr_t)&ldsA[0][ldr * LDSTR + ldk];
    const uint32_t off1 = (uint32_t)(uintptr_t)&ldsA[1][ldr * LDSTR + ldk];

    v8f acc = {};
    const float* bBase = Bswz + (size_t)wave * 64 + lane * 2;

    // Prologue: tiles 0 and 1 in flight.
    async_copy_b128(off0, gsrc);
    async_copy_b128(off1, gsrc + 64);

    for (int kt = 0; kt < SEQ; kt += 64) {
        // Tile kt/64 done when at most the next tile's copy is outstanding.
        if (kt + 64 < SEQ) asm volatile("s_wait_asynccnt 0x1" ::: "memory");
        else               asm volatile("s_wait_asynccnt 0x0" ::: "memory");
        __syncthreads();

        const float* lbuf = &ldsA[(kt >> 6) & 1][0];
        const float* bp   = bBase + (size_t)(kt >> 2) * (NTILE * 64);
#pragma unroll
        for (int kk = 0; kk < 64; kk += 4) {
            v2f af;
            const float* as = &lbuf[row16 * LDSTR + kk + chalf];
            af[0] = as[0];               // A frag: VGPR0=K(c), VGPR1=K(c+1)
            af[1] = as[1];
            v2f bf = *(const v2f*)(bp + (kk >> 2) * (NTILE * 64));
            acc = __builtin_amdgcn_wmma_f32_16x16x4_f32(false, af, false, bf,
                                                        (short)0, acc, false, false);
        }
        __syncthreads();                 // all waves finished reading this buffer

        if (kt + 128 < SEQ) {            // refill it with tile kt/64 + 2
            async_copy_b128(((kt >> 6) & 1) ? off1 : off0, gsrc + kt + 128);
        }
    }

    // C/D layout: VGPR v -> M = v (lanes 0-15) / v+8 (lanes 16-31), N = lane&15.
    const int n = wave * 16 + row16;
    if (n < EMB) {
        const float bv = bias[n];
        const int r0 = m0 + ((lane < 16) ? 0 : 8);
#pragma unroll
        for (int v = 0; v < 8; ++v)
            out[(size_t)(r0 + v) * EMB + n] = fmaxf(acc[v] + bv, 0.f);
    }
}

// ---------------------------------------------------------------------------
// One tree level: hout[nOut x 100] = relu(hin_pairs[nOut x 200] @ W.T + b)
// pairs row r == contiguous hin[200r .. 200r+199] (adjacent sibling rows).
// Whole 16x200 A panel fits in LDS; 50 fully-unrolled WMMA k-steps.
// ---------------------------------------------------------------------------
__global__ __launch_bounds__(256) void level_gemm(const float* __restrict__ hin,
                                                  const float* __restrict__ Wswz,
                                                  const float* __restrict__ bias,
                                                  float* __restrict__ hout,
                                                  int nOut) {
    __shared__ float ldsA[16 * 204];
    const int m0 = blockIdx.x * 16;
    for (int i = threadIdx.x; i < 16 * KLVL; i += 256) {
        int r = i / KLVL;
        int k = i - r * KLVL;
        float v = 0.f;
        if (m0 + r < nOut) v = hin[(size_t)(m0 + r) * KLVL + k];
        ldsA[r * 204 + k] = v;
    }
    __syncthreads();

    const int lane  = threadIdx.x & 31;
    const int wave  = threadIdx.x >> 5;
    const int row16 = lane & 15;
    const int chalf = (lane < 16) ? 0 : 2;

    v8f acc = {};
    const float* bp = Wswz + (size_t)wave * 64 + lane * 2;
#pragma unroll
    for (int ks = 0; ks < KSTEPS_LVL; ++ks) {
        v2f af;
        const float* as = &ldsA[row16 * 204 + ks * 4 + chalf];
        af[0] = as[0];
        af[1] = as[1];
        v2f bf = *(const v2f*)(bp + (size_t)ks * (NTILE * 64));
        acc = __builtin_amdgcn_wmma_f32_16x16x4_f32(false, af, false, bf,
                                                    (short)0, acc, false, false);
    }

    const int n = wave * 16 + row16;
    if (n < EMB) {
        const float bv = bias[n];
        const int r0 = m0 + ((lane < 16) ? 0 : 8);
#pragma unroll
        for (int v = 0; v < 8; ++v) {
            int r = r0 + v;
            if (r < nOut) hout[(size_t)r * EMB + n] = fmaxf(acc[v] + bv, 0.f);
        }
    }
}

// ---------------------------------------------------------------------------
// Root projection: logits[2] = h[100] @ Wp.T + bp   (one wave32)
// ---------------------------------------------------------------------------
__global__ void root_proj(const float* __restrict__ h, const float* __restrict__ Wp,
                          const float* __restrict__ bp, float* __restrict__ out) {
    const int lane = threadIdx.x;        // 32 threads
    float s0 = 0.f, s1 = 0.f;
    for (int k = lane; k < EMB; k += 32) {
        float hv = h[k];
        s0 += hv * Wp[k];
        s1 += hv * Wp[EMB + k];
    }
    for (int off = 16; off > 0; off >>= 1) {
        s0 += __shfl_xor(s0, off, 32);
        s1 += __shfl_xor(s1, off, 32);
    }
    if (lane == 0) {
        out[0] = s0 + bp[0];
        out[1] = s1 + bp[1];
    }
}

// ---------------------------------------------------------------------------
extern "C" void kernel_launch(void* const* d_in, const int* in_sizes, int n_in,
                              void* d_out, int out_size, void* d_ws, size_t ws_size,
                              hipStream_t stream) {
    (void)in_sizes; (void)n_in; (void)out_size; (void)ws_size;
    const float* leaf = (const float*)d_in[0];   // [8192, 4096]
    const float* We   = (const float*)d_in[1];   // [100, 4096]
    const float* be   = (const float*)d_in[2];   // [100]
    const float* W    = (const float*)d_in[3];   // [100, 200]
    const float* b    = (const float*)d_in[4];   // [100]
    const float* Wp   = (const float*)d_in[5];   // [2, 100]
    const float* bp   = (const float*)d_in[6];   // [2]

    // Workspace layout (floats): ~7.2 MB total.
    float* ws     = (float*)d_ws;
    float* BeSwz  = ws;                                       // 1024*8*64
    float* WlSwz  = BeSwz + (size_t)1024 * NTILE * 64;        // 50*8*64
    float* hA     = WlSwz + (size_t)KSTEPS_LVL * NTILE * 64;  // 8192*100
    float* hB     = hA + (size_t)NLEAF * EMB;                 // 4096*100

    // 1) Pre-swizzle both B operands into WMMA fragment layout (L2-resident).
    {
        int frags = (SEQ / 4) * NTILE;           // 8192
        swz_kernel<<<(frags * 32 + 255) / 256, 256, 0, stream>>>(We, BeSwz, EMB, SEQ, frags);
    }
    {
        int frags = KSTEPS_LVL * NTILE;          // 400
        swz_kernel<<<(frags * 32 + 255) / 256, 256, 0, stream>>>(W, WlSwz, EMB, KLVL, frags);
    }

    // 2) Leaf embedding GEMM (memory-bound: streams 128 MiB of leaf_seqs once).
    embed_gemm<<<NLEAF / 16, 256, 0, stream>>>(leaf, BeSwz, be, hA);

    // 3) 13 tree levels, ping-ponging h buffers.
    float* src = hA;
    float* dst = hB;
    int rows = NLEAF;
    for (int lvl = 0; lvl < 13; ++lvl) {
        int nOut = rows >> 1;
        level_gemm<<<(nOut + 15) / 16, 256, 0, stream>>>(src, WlSwz, b, dst, nOut);
        float* t = src; src = dst; dst = t;
        rows = nOut;
    }

    // 4) Root projection -> logits[2].
    root_proj<<<1, 32, 0, stream>>>(src, Wp, bp, (float*)d_out);
}